// UniGraph2_43198781063537
// MI455X (gfx1250) — compile-verified
//
#include <hip/hip_runtime.h>
#include <hip/hip_bf16.h>

#define Nn   6144
#define Ee   98304
#define Hh   256
#define HEADS 4
#define NEx  8
#define Ss   400
#define Ll   3
#define Vv   1024
#define Tt   768

typedef __attribute__((ext_vector_type(2))) float v2f;
typedef __attribute__((ext_vector_type(8))) float v8f;

__device__ __forceinline__ float gelu_exact(float x) {
    return 0.5f * x * (1.0f + erff(x * 0.7071067811865475f));
}

__device__ __forceinline__ float block_reduce_sum_256(float v, float* sh) {
    int t = threadIdx.x;
    sh[t] = v; __syncthreads();
    #pragma unroll
    for (int s = 128; s > 0; s >>= 1) {
        if (t < s) sh[t] += sh[t + s];
        __syncthreads();
    }
    float r = sh[0];
    __syncthreads();
    return r;
}

// ---------------------------------------------------------------------------
// Full-tile fp32 WMMA GEMM: C[M,Nc] = A[M,K] @ B[K,Nc] (+bias), row-major.
// One wave owns a 16x64 output strip: 4 accumulator tiles share one A fragment
// per K-step (V_WMMA_F32_16X16X4_F32). REQUIRES: M%16==0, K%4==0, Nc%64==0.
// LDB is a template constant (256/768/1024) so all B loads are immediate
// offsets from one running pointer -> single s_clause per K-step, no 64-bit
// address math in the loop.  EPI==0: C = acc+bias.
// EPI==1: C += rowscale[m*rs_stride]*(acc+bias).
// ---------------------------------------------------------------------------
template<int EPI, int LDB>
__global__ __launch_bounds__(32) void wmma_gemm(
    const float* __restrict__ A, int lda,
    const float* __restrict__ B,
    const float* __restrict__ bias,
    float* __restrict__ C, int ldc,
    int K,
    const float* __restrict__ rowscale, int rs_stride)
{
    const int tileM  = blockIdx.x << 4;
    const int tileN0 = blockIdx.y << 6;            // 64-wide strip
    const int lane   = threadIdx.x & 31;
    const int half   = lane >> 4;                  // 0: rows K0/K1, 1: rows K2/K3
    const int l16    = lane & 15;

    const float* __restrict__ Ap = A + (size_t)(tileM + l16) * lda + (half << 1);
    const float* __restrict__ Bp = B + (size_t)(half << 1) * LDB + tileN0 + l16;

    v8f acc[4];
    #pragma unroll
    for (int t = 0; t < 4; ++t) acc[t] = (v8f){};

    #pragma unroll 2
    for (int k0 = 0; k0 < K; k0 += 4) {
        v2f a = *(const v2f*)Ap;                    // contiguous pair -> b64 load
        v2f b0, b1, b2, b3;                         // 8 loads, immediate offsets
        b0.x = Bp[0];        b0.y = Bp[LDB];
        b1.x = Bp[16];       b1.y = Bp[LDB + 16];
        b2.x = Bp[32];       b2.y = Bp[LDB + 32];
        b3.x = Bp[48];       b3.y = Bp[LDB + 48];
        acc[0] = __builtin_amdgcn_wmma_f32_16x16x4_f32(false, a, false, b0, (short)0, acc[0], false, false);
        acc[1] = __builtin_amdgcn_wmma_f32_16x16x4_f32(false, a, false, b1, (short)0, acc[1], false, false);
        acc[2] = __builtin_amdgcn_wmma_f32_16x16x4_f32(false, a, false, b2, (short)0, acc[2], false, false);
        acc[3] = __builtin_amdgcn_wmma_f32_16x16x4_f32(false, a, false, b3, (short)0, acc[3], false, false);
        Ap += 4;
        Bp += 4 * LDB;
    }

    #pragma unroll
    for (int r = 0; r < 8; ++r) {
        const int m = tileM + r + (half << 3);      // lanes 16-31 hold rows +8
        float* Crow = C + (size_t)m * ldc + tileN0 + l16;
        const float rs = (EPI == 1) ? rowscale[(size_t)m * rs_stride] : 0.0f;
        #pragma unroll
        for (int t = 0; t < 4; ++t) {
            const int col = tileN0 + (t << 4) + l16;
            float v = acc[t][r] + (bias ? bias[col] : 0.0f);
            if (EPI == 0) Crow[t << 4] = v;
            else          Crow[t << 4] += rs * v;
        }
    }
}

// ---------------------------------------------------------------------------
// Elementwise / reduction kernels
// ---------------------------------------------------------------------------
__global__ void combine_proj(const float* __restrict__ a, const float* __restrict__ b,
                             const int* __restrict__ mask_int,
                             const float* __restrict__ mask_token,
                             float* __restrict__ x)
{
    int n = blockIdx.x, k = threadIdx.x;
    size_t idx = (size_t)n * Hh + k;
    float v = 0.5f * (a[idx] + b[idx]);
    if (mask_int[n] == 0) v = mask_token[k];
    x[idx] = v;
}

__global__ void mask_count(const int* __restrict__ mask_int, float* __restrict__ acc)
{
    __shared__ float sh[256];
    int t = blockIdx.x * 256 + threadIdx.x;
    float v = (t < Nn && mask_int[t] == 0) ? 1.0f : 0.0f;
    v = block_reduce_sum_256(v, sh);
    if (threadIdx.x == 0) atomicAdd(acc, v);
}

// tiny gate projection (Nc=8): plain VALU, avoids partial WMMA tiles
__global__ void gate_logits(const float* __restrict__ x, const float* __restrict__ gw,
                            const float* __restrict__ gb, float* __restrict__ out)
{
    int t = blockIdx.x * blockDim.x + threadIdx.x;
    if (t >= Nn * NEx) return;
    int n = t / NEx, e = t % NEx;
    const float* xr = x + (size_t)n * Hh;
    float s = gb[e];
    for (int k = 0; k < Hh; ++k) s += xr[k] * gw[k * NEx + e];
    out[t] = s;
}

__global__ void gate_topk(const float* __restrict__ logits, float* __restrict__ wsel)
{
    int n = blockIdx.x * blockDim.x + threadIdx.x;
    if (n >= Nn) return;
    const float* lg = logits + (size_t)n * NEx;
    float p[NEx];
    float mx = lg[0];
    #pragma unroll
    for (int i = 1; i < NEx; ++i) mx = fmaxf(mx, lg[i]);
    float s = 0.0f;
    #pragma unroll
    for (int i = 0; i < NEx; ++i) { p[i] = expf(lg[i] - mx); s += p[i]; }
    #pragma unroll
    for (int i = 0; i < NEx; ++i) p[i] /= s;
    int i0 = 0;
    #pragma unroll
    for (int i = 1; i < NEx; ++i) if (p[i] > p[i0]) i0 = i;
    int i1 = (i0 == 0) ? 1 : 0;
    #pragma unroll
    for (int i = 0; i < NEx; ++i) if (i != i0 && p[i] > p[i1]) i1 = i;
    float sw = p[i0] + p[i1];
    #pragma unroll
    for (int i = 0; i < NEx; ++i) wsel[(size_t)n * NEx + i] = 0.0f;
    wsel[(size_t)n * NEx + i0] = p[i0] / sw;
    wsel[(size_t)n * NEx + i1] = p[i1] / sw;
}

__global__ void ln_gelu(float* __restrict__ X, const float* __restrict__ g,
                        const float* __restrict__ b)
{
    __shared__ float sh[256];
    int n = blockIdx.x, k = threadIdx.x;
    size_t idx = (size_t)n * Hh + k;
    float v = X[idx];
    float mean = block_reduce_sum_256(v, sh) * (1.0f / Hh);
    float d = v - mean;
    float var = block_reduce_sum_256(d * d, sh) * (1.0f / Hh);
    float y = d * rsqrtf(var + 1e-5f) * g[k] + b[k];
    X[idx] = gelu_exact(y);
}

// --- GAT attention helpers -------------------------------------------------
__global__ void compute_elr(const float* __restrict__ z,
                            const float* __restrict__ al, const float* __restrict__ ar,
                            float* __restrict__ el, float* __restrict__ er)
{
    int t = blockIdx.x * blockDim.x + threadIdx.x;
    if (t >= Nn * HEADS) return;
    int hd = t % HEADS;
    const float* zr = z + (size_t)t * Hh;       // t = n*HEADS + hd; z is [n][hd][k]
    const float* a  = al + hd * Hh;
    const float* r  = ar + hd * Hh;
    float sl = 0.0f, sr = 0.0f;
    for (int k = 0; k < Hh; ++k) { float zv = zr[k]; sl += zv * a[k]; sr += zv * r[k]; }
    el[t] = sl; er[t] = sr;
}

__device__ __forceinline__ unsigned f2ord(float f) {
    unsigned u = __float_as_uint(f);
    return (u & 0x80000000u) ? ~u : (u | 0x80000000u);
}
__device__ __forceinline__ float ord2f(unsigned u) {
    return (u & 0x80000000u) ? __uint_as_float(u ^ 0x80000000u) : __uint_as_float(~u);
}

__global__ void edge_max(const int* __restrict__ src, const int* __restrict__ dst,
                         const float* __restrict__ el, const float* __restrict__ er,
                         unsigned* __restrict__ mb)
{
    int t = blockIdx.x * blockDim.x + threadIdx.x;
    if (t >= Ee * HEADS) return;
    int e = t / HEADS, hd = t % HEADS;
    int s = src[e], d = dst[e];
    float v = el[s * HEADS + hd] + er[d * HEADS + hd];
    v = v > 0.0f ? v : 0.2f * v;                 // leaky_relu(0.2)
    atomicMax(&mb[d * HEADS + hd], f2ord(v));
}

__global__ void edge_expsum(const int* __restrict__ src, const int* __restrict__ dst,
                            const float* __restrict__ el, const float* __restrict__ er,
                            const unsigned* __restrict__ mb,
                            float* __restrict__ den, float* __restrict__ ex)
{
    int t = blockIdx.x * blockDim.x + threadIdx.x;
    if (t >= Ee * HEADS) return;
    int e = t / HEADS, hd = t % HEADS;
    int s = src[e], d = dst[e];
    float v = el[s * HEADS + hd] + er[d * HEADS + hd];
    v = v > 0.0f ? v : 0.2f * v;
    float mm  = ord2f(mb[d * HEADS + hd]);
    float exv = expf(v - mm);
    ex[t] = exv;
    atomicAdd(&den[d * HEADS + hd], exv);
}

__global__ void edge_agg(const int* __restrict__ src, const int* __restrict__ dst,
                         const float* __restrict__ ex, const float* __restrict__ den,
                         const float* __restrict__ z, float* __restrict__ out)
{
    int e = blockIdx.x, k = threadIdx.x;
    int s = src[e], d = dst[e];
    #pragma unroll
    for (int hd = 0; hd < HEADS; ++hd) {
        float dn = den[d * HEADS + hd];
        dn = dn > 0.0f ? dn : 1.0f;
        float alpha = ex[(size_t)e * HEADS + hd] / dn;
        atomicAdd(&out[((size_t)d * HEADS + hd) * Hh + k],
                  alpha * z[((size_t)s * HEADS + hd) * Hh + k]);
    }
}

__global__ void gat_finalize(const float* __restrict__ out, const float* __restrict__ bias,
                             float* __restrict__ h)
{
    int n = blockIdx.x, k = threadIdx.x;
    float s = 0.0f;
    #pragma unroll
    for (int hd = 0; hd < HEADS; ++hd)
        s += out[((size_t)n * HEADS + hd) * Hh + k] + bias[hd * Hh + k];
    h[(size_t)n * Hh + k] = s * 0.25f;
}

// --- decoder focal-cosine loss --------------------------------------------
__global__ void cos_loss(const float* __restrict__ r, const float* __restrict__ tgt,
                         int D, const int* __restrict__ mask_int, float* __restrict__ acc)
{
    __shared__ float sh[256];
    int n = blockIdx.x, t = threadIdx.x;
    float sab = 0.0f, saa = 0.0f, sbb = 0.0f;
    for (int k = t; k < D; k += 256) {
        float a = r[(size_t)n * D + k], bv = tgt[(size_t)n * D + k];
        sab += a * bv; saa += a * a; sbb += bv * bv;
    }
    sab = block_reduce_sum_256(sab, sh);
    saa = block_reduce_sum_256(saa, sh);
    sbb = block_reduce_sum_256(sbb, sh);
    if (t == 0 && mask_int[n] == 0) {
        float na = fmaxf(sqrtf(saa), 1e-8f);
        float nb = fmaxf(sqrtf(sbb), 1e-8f);
        float sim = sab / (na * nb);
        float om = 1.0f - sim;
        atomicAdd(acc, om * om);            // GAMMA = 2
    }
}

// --- SPD pairwise decode ---------------------------------------------------
__global__ void gather_rows(const float* __restrict__ h, const int* __restrict__ sidx,
                            float* __restrict__ se)
{
    int i = blockIdx.x, k = threadIdx.x;
    se[(size_t)i * Hh + k] = h[(size_t)sidx[i] * Hh + k];
}

__global__ void spd_pair(const float* __restrict__ li, const float* __restrict__ lj,
                         const float* __restrict__ b1,
                         const float* __restrict__ g, const float* __restrict__ b,
                         const float* __restrict__ w2, const float* __restrict__ b2,
                         const float* __restrict__ spd, const int* __restrict__ sidx,
                         float* __restrict__ acc)
{
    __shared__ float sh[256];
    int i = blockIdx.x, j = blockIdx.y, k = threadIdx.x;
    float pre = li[(size_t)i * Hh + k] + lj[(size_t)j * Hh + k] + b1[k];
    float mean = block_reduce_sum_256(pre, sh) * (1.0f / Hh);
    float d = pre - mean;
    float var = block_reduce_sum_256(d * d, sh) * (1.0f / Hh);
    float y = d * rsqrtf(var + 1e-5f) * g[k] + b[k];
    float contrib = gelu_exact(y) * w2[k];
    float pred = block_reduce_sum_256(contrib, sh) + b2[0];
    if (k == 0) {
        float st = spd[(size_t)sidx[i] * Nn + sidx[j]];
        float df = pred - st;
        atomicAdd(acc, df * df);
    }
}

__global__ void finalize_loss(const float* __restrict__ acc, float* __restrict__ out)
{
    float nm = fmaxf(acc[3], 1.0f);
    out[0] = acc[0] / nm + acc[1] / nm + 0.5f * (acc[2] / (float)(Ss * Ss));
}

__global__ void copy_out(const float* __restrict__ h, float* __restrict__ out)
{
    int t = blockIdx.x * blockDim.x + threadIdx.x;
    if (t < Nn * Hh) out[t] = h[t];
}

// ---------------------------------------------------------------------------
template<int LDB>
static inline void gemm0(const float* A, int lda, const float* B,
                         const float* bias, float* C, int ldc,
                         int M, int K, int Nc, hipStream_t stream)
{
    dim3 grid(M / 16, Nc / 64);
    wmma_gemm<0, LDB><<<grid, dim3(32), 0, stream>>>(A, lda, B, bias, C, ldc,
                                                     K, (const float*)nullptr, 0);
}

extern "C" void kernel_launch(void* const* d_in, const int* in_sizes, int n_in,
                              void* d_out, int out_size, void* d_ws, size_t ws_size,
                              hipStream_t stream)
{
    (void)in_sizes; (void)n_in; (void)out_size; (void)ws_size;
    const float* feat_v     = (const float*)d_in[0];
    const float* feat_t     = (const float*)d_in[1];
    const float* spd_matrix = (const float*)d_in[2];
    const float* proj_v_w   = (const float*)d_in[3];
    const float* proj_v_b   = (const float*)d_in[4];
    const float* proj_t_w   = (const float*)d_in[5];
    const float* proj_t_b   = (const float*)d_in[6];
    const float* mask_token = (const float*)d_in[7];
    const float* gate_w     = (const float*)d_in[8];
    const float* gate_b     = (const float*)d_in[9];
    const float* moe_w1     = (const float*)d_in[10];
    const float* moe_b1     = (const float*)d_in[11];
    const float* moe_ln_g   = (const float*)d_in[12];
    const float* moe_ln_b   = (const float*)d_in[13];
    const float* moe_w2     = (const float*)d_in[14];
    const float* moe_b2     = (const float*)d_in[15];
    const float* gat_fc     = (const float*)d_in[16];
    const float* gat_al     = (const float*)d_in[17];
    const float* gat_ar     = (const float*)d_in[18];
    const float* gat_bias   = (const float*)d_in[19];
    const float* dec_v_w1   = (const float*)d_in[20];
    const float* dec_v_b1   = (const float*)d_in[21];
    const float* dec_v_ln_g = (const float*)d_in[22];
    const float* dec_v_ln_b = (const float*)d_in[23];
    const float* dec_v_w2   = (const float*)d_in[24];
    const float* dec_v_b2   = (const float*)d_in[25];
    const float* dec_t_w1   = (const float*)d_in[26];
    const float* dec_t_b1   = (const float*)d_in[27];
    const float* dec_t_ln_g = (const float*)d_in[28];
    const float* dec_t_ln_b = (const float*)d_in[29];
    const float* dec_t_w2   = (const float*)d_in[30];
    const float* dec_t_b2   = (const float*)d_in[31];
    const float* spd_w1     = (const float*)d_in[32];
    const float* spd_b1     = (const float*)d_in[33];
    const float* spd_ln_g   = (const float*)d_in[34];
    const float* spd_ln_b   = (const float*)d_in[35];
    const float* spd_w2     = (const float*)d_in[36];
    const float* spd_b2     = (const float*)d_in[37];
    const int*   src        = (const int*)d_in[38];
    const int*   dst        = (const int*)d_in[39];
    const int*   mask_int   = (const int*)d_in[40];
    const int*   sample_idx = (const int*)d_in[41];

    // ---- workspace carve-up (floats) ----
    float* W = (float*)d_ws;
    size_t off = 0;
    auto take = [&](size_t n) { float* p = W + off; off += (n + 63) & ~(size_t)63; return p; };
    float* xb   = take((size_t)Nn * Hh);            // projected+masked features
    float* hb   = take((size_t)Nn * Hh);            // running node embedding
    float* bufA = take((size_t)Nn * HEADS * Hh);    // proj_v tmp / GAT z / decoder r
    float* bufB = take((size_t)Nn * HEADS * Hh);    // proj_t tmp / expert hidden / GAT out / dec t
    float* wsel = take((size_t)Nn * NEx);           // sparse top-2 gate weights
    float* elb  = take((size_t)Nn * HEADS);
    float* erb  = take((size_t)Nn * HEADS);
    float* mb   = take((size_t)Nn * HEADS);         // ordered-uint segment max
    float* denb = take((size_t)Nn * HEADS);
    float* exb  = take((size_t)Ee * HEADS);
    float* seb  = take((size_t)Ss * Hh);
    float* lib  = take((size_t)Ss * Hh);
    float* ljb  = take((size_t)Ss * Hh);
    float* accb = take(64);                         // [recV, recT, spd, mask_count]

    hipMemsetAsync(accb, 0, 64 * sizeof(float), stream);

    // ---- projections + mask token ----
    gemm0<Hh>(feat_v, Vv, proj_v_w, proj_v_b, bufA, Hh, Nn, Vv, Hh, stream);
    gemm0<Hh>(feat_t, Tt, proj_t_w, proj_t_b, bufB, Hh, Nn, Tt, Hh, stream);
    combine_proj<<<Nn, Hh, 0, stream>>>(bufA, bufB, mask_int, mask_token, xb);
    mask_count<<<(Nn + 255) / 256, 256, 0, stream>>>(mask_int, accb + 3);

    // ---- MoE: gate softmax + top-2, dense experts fused via rowscale GEMM ----
    gate_logits<<<(Nn * NEx + 255) / 256, 256, 0, stream>>>(xb, gate_w, gate_b, bufA);
    gate_topk<<<(Nn + 255) / 256, 256, 0, stream>>>(bufA, wsel);
    hipMemsetAsync(hb, 0, (size_t)Nn * Hh * sizeof(float), stream);
    for (int e = 0; e < NEx; ++e) {
        gemm0<Hh>(xb, Hh, moe_w1 + (size_t)e * Hh * Hh, moe_b1 + e * Hh,
                  bufB, Hh, Nn, Hh, Hh, stream);
        ln_gelu<<<Nn, Hh, 0, stream>>>(bufB, moe_ln_g + e * Hh, moe_ln_b + e * Hh);
        dim3 grid(Nn / 16, Hh / 64);
        wmma_gemm<1, Hh><<<grid, dim3(32), 0, stream>>>(
            bufB, Hh, moe_w2 + (size_t)e * Hh * Hh, moe_b2 + e * Hh,
            hb, Hh, Hh, wsel + e, NEx);
    }

    // ---- GAT layers ----
    for (int l = 0; l < Ll; ++l) {
        gemm0<HEADS * Hh>(hb, Hh, gat_fc + (size_t)l * Hh * HEADS * Hh, nullptr,
                          bufA, HEADS * Hh, Nn, Hh, HEADS * Hh, stream);
        compute_elr<<<(Nn * HEADS + 255) / 256, 256, 0, stream>>>(
            bufA, gat_al + (size_t)l * HEADS * Hh, gat_ar + (size_t)l * HEADS * Hh, elb, erb);
        hipMemsetAsync(mb, 0, (size_t)Nn * HEADS * sizeof(float), stream);
        hipMemsetAsync(denb, 0, (size_t)Nn * HEADS * sizeof(float), stream);
        edge_max<<<(Ee * HEADS + 255) / 256, 256, 0, stream>>>(src, dst, elb, erb, (unsigned*)mb);
        edge_expsum<<<(Ee * HEADS + 255) / 256, 256, 0, stream>>>(src, dst, elb, erb,
                                                                  (const unsigned*)mb, denb, exb);
        hipMemsetAsync(bufB, 0, (size_t)Nn * HEADS * Hh * sizeof(float), stream);
        edge_agg<<<Ee, Hh, 0, stream>>>(src, dst, exb, denb, bufA, bufB);
        gat_finalize<<<Nn, Hh, 0, stream>>>(bufB, gat_bias + (size_t)l * HEADS * Hh, hb);
    }

    // ---- decoders + focal cosine reconstruction ----
    gemm0<Hh>(hb, Hh, dec_v_w1, dec_v_b1, bufB, Hh, Nn, Hh, Hh, stream);
    ln_gelu<<<Nn, Hh, 0, stream>>>(bufB, dec_v_ln_g, dec_v_ln_b);
    gemm0<Vv>(bufB, Hh, dec_v_w2, dec_v_b2, bufA, Vv, Nn, Hh, Vv, stream);
    cos_loss<<<Nn, 256, 0, stream>>>(bufA, feat_v, Vv, mask_int, accb + 0);

    gemm0<Hh>(hb, Hh, dec_t_w1, dec_t_b1, bufB, Hh, Nn, Hh, Hh, stream);
    ln_gelu<<<Nn, Hh, 0, stream>>>(bufB, dec_t_ln_g, dec_t_ln_b);
    gemm0<Tt>(bufB, Hh, dec_t_w2, dec_t_b2, bufA, Tt, Nn, Hh, Tt, stream);
    cos_loss<<<Nn, 256, 0, stream>>>(bufA, feat_t, Tt, mask_int, accb + 1);

    // ---- SPD pairwise loss ----
    gather_rows<<<Ss, Hh, 0, stream>>>(hb, sample_idx, seb);
    gemm0<Hh>(seb, Hh, spd_w1,           nullptr, lib, Hh, Ss, Hh, Hh, stream);
    gemm0<Hh>(seb, Hh, spd_w1 + Hh * Hh, nullptr, ljb, Hh, Ss, Hh, Hh, stream);
    spd_pair<<<dim3(Ss, Ss), Hh, 0, stream>>>(lib, ljb, spd_b1, spd_ln_g, spd_ln_b,
                                              spd_w2, spd_b2, spd_matrix, sample_idx, accb + 2);

    // ---- outputs: [total_loss, h flat] ----
    float* out = (float*)d_out;
    finalize_loss<<<1, 1, 0, stream>>>(accb, out);
    copy_out<<<(Nn * Hh + 255) / 256, 256, 0, stream>>>(hb, out + 1);
}